// NPairLoss_13254269075869
// MI455X (gfx1250) — compile-verified
//
#include <hip/hip_runtime.h>
#include <hip/hip_bf16.h>

// N-pair contrastive loss, fused. B=8192, D=256, T=0.1.
// Gram via V_WMMA_F32_16X16X4_F32 (fp32 matrix pipe, wave32), fully fused
// with masked online logsumexp (negatives) and Gumbel-argmax positive pick,
// so the 8192x8192 sim matrix is never materialized (saves ~0.5 GB traffic
// and L2 capacity; embeddings (8 MB) stay L2-resident on the 192 MB L2).

#define B_N 8192
#define D_K 256
#define INV_T 10.0f
#define ROWS_PER_BLOCK 16
#define WAVES 8
#define LDS_STRIDE 260   // 260 % 64 == 4 -> 16 rows hit 16 distinct LDS banks

typedef __attribute__((ext_vector_type(2))) float v2f;
typedef __attribute__((ext_vector_type(8))) float v8f;

// Stateless per-(i,j) Gumbel noise (murmur-style finalizer). Stand-in for
// jax.random.gumbel(key=42): structurally identical (uniform iid -> gumbel).
__device__ __forceinline__ float gumbel_noise(unsigned i, unsigned j) {
    unsigned x = i * 8192u + j + 0x2545F491u;
    x ^= x >> 16; x *= 0x7feb352du;
    x ^= x >> 15; x *= 0x846ca68bu;
    x ^= x >> 16;
    float u = ((x >> 8) + 0.5f) * (1.0f / 16777216.0f); // (0,1)
    return -__logf(-__logf(u));
}

// Kernel 1: per-row squared L2 norms + label narrowing (int64 -> int32).
__global__ void npair_rownorm_kernel(const float* __restrict__ emb,
                                     const long long* __restrict__ lab64,
                                     float* __restrict__ sq,
                                     int* __restrict__ lab) {
    const int row  = blockIdx.x * 8 + (threadIdx.x >> 5);
    const int lane = threadIdx.x & 31;
    const float* r = emb + (size_t)row * D_K;
    float s = 0.f;
    #pragma unroll
    for (int c = lane; c < D_K; c += 32) { float v = r[c]; s += v * v; }
    #pragma unroll
    for (int off = 16; off; off >>= 1) s += __shfl_xor(s, off, 32);
    if (lane == 0) { sq[row] = s; lab[row] = (int)lab64[row]; }
}

// Kernel 2: fused Gram + masked logsumexp + Gumbel-argmax positive.
// Block = 8 waves, owns a 16-row strip; each wave sweeps 16-column tiles.
__global__ void __launch_bounds__(256, 2)
npair_main_kernel(const float* __restrict__ emb,
                  const float* __restrict__ sq,
                  const int* __restrict__ lab,
                  float* __restrict__ acc) {
    __shared__ float sA[ROWS_PER_BLOCK * LDS_STRIDE];
    __shared__ float sSq[ROWS_PER_BLOCK];
    __shared__ int   sLab[ROWS_PER_BLOCK];
    __shared__ float sRed[WAVES][ROWS_PER_BLOCK][4];

    const int tid  = threadIdx.x;
    const int wave = tid >> 5;
    const int lane = tid & 31;
    const int hi   = lane >> 4;   // half-wave select
    const int lm   = lane & 15;
    const int rowBase = blockIdx.x * ROWS_PER_BLOCK;

    // Stage the 16xD A-strip into LDS (coalesced), padded stride.
    for (int idx = tid; idx < ROWS_PER_BLOCK * D_K; idx += 256) {
        const int r = idx >> 8, c = idx & (D_K - 1);
        sA[r * LDS_STRIDE + c] = emb[(size_t)(rowBase + r) * D_K + c];
    }
    if (tid < ROWS_PER_BLOCK) {
        sSq[tid]  = sq[rowBase + tid];
        sLab[tid] = lab[rowBase + tid];
    }
    __syncthreads();

    // Per-lane online accumulators: v indexes C-tile VGPR -> row m = 8*hi + v.
    float runMax[8], runSum[8], bestG[8], bestSim[8];
    float sqi[8]; int labi[8];
    const int mrow = hi << 3;
    #pragma unroll
    for (int v = 0; v < 8; ++v) {
        runMax[v] = -3.0e38f; runSum[v] = 0.f;
        bestG[v]  = -3.0e38f; bestSim[v] = 0.f;
        sqi[v] = sSq[mrow + v]; labi[v] = sLab[mrow + v];
    }

    const float* aptr = &sA[lm * LDS_STRIDE + 2 * hi];

    for (int colBase = wave * 16; colBase < B_N; colBase += WAVES * 16) {
        const int   n    = colBase + lm;          // this lane's column
        const float sqn  = sq[n];
        const int   labn = lab[n];
        const float* bptr = emb + (size_t)n * D_K + 2 * hi;
        // Prefetch next tile's column row (stays non-speculative in L2).
        if (n + WAVES * 16 < B_N)
            __builtin_prefetch(emb + (size_t)(n + WAVES * 16) * D_K, 0, 0);

        v8f c = {0.f, 0.f, 0.f, 0.f, 0.f, 0.f, 0.f, 0.f};
        // 16x16 Gram tile: 64 chained fp32 WMMAs over K=256.
        // A lane layout: row=lm, K pair = k + 2*hi (+0,+1).
        // B = E^T of the same matrix -> identical per-lane layout on column n.
        #pragma unroll
        for (int k = 0; k < D_K; k += 4) {
            v2f a = *(const v2f*)(aptr + k);
            v2f b = *(const v2f*)(bptr + k);
            c = __builtin_amdgcn_wmma_f32_16x16x4_f32(
                    false, a, false, b, (short)0, c, false, false);
        }

        #pragma unroll
        for (int v = 0; v < 8; ++v) {
            const int i = rowBase + mrow + v;
            // sim = -max(dist2,0)/T = min((2g - |xi|^2 - |xj|^2) * 10, 0)
            const float simv = fminf((2.f * c[v] - sqi[v] - sqn) * INV_T, 0.f);
            if (labi[v] != labn) {                 // negative: online logsumexp
                const float mnew = fmaxf(runMax[v], simv);
                runSum[v] = runSum[v] * __expf(runMax[v] - mnew)
                          + __expf(simv - mnew);
                runMax[v] = mnew;
            } else if (i != n) {                   // positive: gumbel argmax
                const float g = gumbel_noise((unsigned)i, (unsigned)n);
                if (g > bestG[v]) { bestG[v] = g; bestSim[v] = simv; }
            }
        }
    }

    // Butterfly reduce across the 16 lanes of each half-wave (masks < 16
    // never cross the half boundary, so row ownership is preserved).
    #pragma unroll
    for (int off = 1; off < 16; off <<= 1) {
        #pragma unroll
        for (int v = 0; v < 8; ++v) {
            const float oM = __shfl_xor(runMax[v], off, 32);
            const float oS = __shfl_xor(runSum[v], off, 32);
            const float mnew = fmaxf(runMax[v], oM);
            runSum[v] = runSum[v] * __expf(runMax[v] - mnew)
                      + oS * __expf(oM - mnew);
            runMax[v] = mnew;
            const float oG = __shfl_xor(bestG[v], off, 32);
            const float oB = __shfl_xor(bestSim[v], off, 32);
            if (oG > bestG[v]) { bestG[v] = oG; bestSim[v] = oB; }
        }
    }
    if (lm == 0) {   // lane 0 -> rows 0..7, lane 16 -> rows 8..15
        #pragma unroll
        for (int v = 0; v < 8; ++v) {
            const int m = mrow + v;
            sRed[wave][m][0] = runMax[v];
            sRed[wave][m][1] = runSum[v];
            sRed[wave][m][2] = bestG[v];
            sRed[wave][m][3] = bestSim[v];
        }
    }
    __syncthreads();

    // Cross-wave merge: one thread per row.
    if (tid < ROWS_PER_BLOCK) {
        float rM = -3.0e38f, rS = 0.f, gB = -3.0e38f, sB = 0.f;
        #pragma unroll
        for (int w = 0; w < WAVES; ++w) {
            const float oM = sRed[w][tid][0], oS = sRed[w][tid][1];
            const float mnew = fmaxf(rM, oM);
            rS = rS * __expf(rM - mnew) + oS * __expf(oM - mnew);
            rM = mnew;
            if (sRed[w][tid][2] > gB) { gB = sRed[w][tid][2]; sB = sRed[w][tid][3]; }
        }
        const bool hasNeg = (rS > 0.f);
        const bool hasPos = (gB > -1.0e38f);
        if (hasNeg && hasPos) {
            const float denom = rM + __logf(rS);
            atomicAdd(&acc[0], denom - sB);  // per_row = -(num - denom)
            atomicAdd(&acc[1], 1.0f);
        }
    }
}

__global__ void npair_finalize_kernel(const float* __restrict__ acc,
                                      float* __restrict__ out) {
    if (threadIdx.x == 0 && blockIdx.x == 0) {
        const float s = acc[0], c = acc[1];
        out[0] = (c > 0.f) ? (s / fmaxf(c, 1.f)) : 0.f;
    }
}

extern "C" void kernel_launch(void* const* d_in, const int* in_sizes, int n_in,
                              void* d_out, int out_size, void* d_ws, size_t ws_size,
                              hipStream_t stream) {
    (void)in_sizes; (void)n_in; (void)out_size; (void)ws_size;
    const float*     emb   = (const float*)d_in[0];
    const long long* lab64 = (const long long*)d_in[1];   // int64 labels

    float* sq  = (float*)d_ws;                                    // B floats
    int*   lab = (int*)((char*)d_ws + (size_t)B_N * sizeof(float));
    float* acc = (float*)((char*)d_ws + (size_t)B_N * (sizeof(float) + sizeof(int)));

    hipMemsetAsync(acc, 0, 2 * sizeof(float), stream);  // deterministic init

    npair_rownorm_kernel<<<B_N / 8, 256, 0, stream>>>(emb, lab64, sq, lab);
    npair_main_kernel<<<B_N / ROWS_PER_BLOCK, 256, 0, stream>>>(emb, sq, lab, acc);
    npair_finalize_kernel<<<1, 64, 0, stream>>>(acc, (float*)d_out);
}